// BlobLoss_17849884082374
// MI455X (gfx1250) — compile-verified
//
#include <hip/hip_runtime.h>
#include <math.h>

#define SIDE 14
#define P 196
#define SS (197 * 197)          // 38809 elements per (b,h) attention map
#define BIGL (1 << 20)          // background sentinel (> any pixel index)

typedef float v2f __attribute__((ext_vector_type(2)));
typedef float v8f __attribute__((ext_vector_type(8)));

typedef __attribute__((address_space(1))) int gint;   // __device__ int*
typedef __attribute__((address_space(3))) int lint;   // __shared__ int*

#if __has_builtin(__builtin_amdgcn_global_load_async_to_lds_b32) && \
    __has_builtin(__builtin_amdgcn_s_wait_asynccnt)
#define USE_ASYNC_LDS 1
#endif

__device__ __forceinline__ int imin(int a, int b) { return a < b ? a : b; }

// ---------------------------------------------------------------------------
// Pass 1: per-row means of the 1536x196 score matrix via WMMA row-sums.
// One wave (32 threads) per 16-row tile; D = A(16x4) * ones(4x16) + C,
// accumulated over 49 K-chunks. D[m, n] == rowsum(m) for every n.
// Out-of-range lanes re-read row 0 (clamped) so loads are unconditional and
// EXEC stays all-ones across the WMMA chain; their rows are never stored.
// ---------------------------------------------------------------------------
__global__ void means_wmma(const float* __restrict__ dq,
                           float* __restrict__ means, int n) {
  const int lane = threadIdx.x;           // 0..31, one wave
  const int tile = blockIdx.x;
  const int m = lane & 15;                // row within tile (A layout)
  const int khalf = (lane >> 4) * 2;      // lanes 0-15: K+{0,1}; 16-31: K+{2,3}
  int row = tile * 16 + m;
  if (row >= n) row = 0;                  // clamp: dead lanes re-read row 0
  const float* rbase = dq + (size_t)row * SS + 1 + khalf;  // [:, :, 0, 1:]

  v2f b; b.x = 1.0f; b.y = 1.0f;          // ones matrix (layout-independent)
  v8f c = {0.f, 0.f, 0.f, 0.f, 0.f, 0.f, 0.f, 0.f};

  for (int k0 = 0; k0 < P; k0 += 4) {     // 196 = 49 * 4, no tail
    v2f a;
    a.x = rbase[k0 + 0];
    a.y = rbase[k0 + 1];
    c = __builtin_amdgcn_wmma_f32_16x16x4_f32(false, a, false, b,
                                              (short)0, c, false, false);
  }

  // c[i]: lanes 0-15 hold M=i, lanes 16-31 hold M=8+i (N = lane&15, all equal)
  __shared__ float rs[16];
  if ((lane & 15) == 0) {
    const int half = lane >> 4;
#pragma unroll
    for (int i = 0; i < 8; ++i) rs[half * 8 + i] = c[i];
  }
  __syncthreads();
  if (lane < 16 && (tile * 16 + lane) < n)
    means[tile * 16 + lane] = rs[lane] * (1.0f / 196.0f);
}

// ---------------------------------------------------------------------------
// Pass 2: one 256-thread block per (b,h) image. Async-load scores to LDS,
// threshold, 196-step 8-connected min-label propagation on a BIG-padded
// 16x16 LDS tile, then deterministic per-component entropy.
// ---------------------------------------------------------------------------
__global__ void blob_kernel(const float* __restrict__ dq,
                            const float* __restrict__ means,
                            float* __restrict__ hsum) {
  __shared__ float xs[256];
  __shared__ float xvs[P];
  __shared__ int   lab[256];   // 16x16, 1-pixel BIG border
  __shared__ float red[256];

  const int t = threadIdx.x;
  const int img = blockIdx.x;
  const float* base = dq + (size_t)img * SS + 1;

#ifdef USE_ASYNC_LDS
  if (t < P) {
    __builtin_amdgcn_global_load_async_to_lds_b32(
        (gint*)(unsigned long long)(base + t), (lint*)(&xs[t]), 0, 0);
  }
  __builtin_amdgcn_s_wait_asynccnt(0);
  __syncthreads();
  const float x = (t < P) ? xs[t] : 0.0f;
#else
  const float x = (t < P) ? base[t] : 0.0f;
#endif

  const float m = means[img];
  const bool msk = (t < P) && (x > m);
  float xvv = 0.0f;
  if (t < P) { xvv = fmaxf(x - m, 0.0f) + 1e-9f; xvs[t] = xvv; }

  lab[t] = BIGL;
  __syncthreads();
  int pos = 0;
  if (t < P) {
    const int r = t / SIDE, cc = t % SIDE;
    pos = (r + 1) * 16 + (cc + 1);
    lab[pos] = msk ? t : BIGL;
  }
  __syncthreads();

  for (int it = 0; it < P; ++it) {        // matches reference's 196-step scan
    int nv = 0;
    if (msk) {
      nv = lab[pos];
      nv = imin(nv, lab[pos - 17]); nv = imin(nv, lab[pos - 16]);
      nv = imin(nv, lab[pos - 15]); nv = imin(nv, lab[pos - 1]);
      nv = imin(nv, lab[pos + 1]);  nv = imin(nv, lab[pos + 15]);
      nv = imin(nv, lab[pos + 16]); nv = imin(nv, lab[pos + 17]);
    }
    __syncthreads();
    if (msk) lab[pos] = nv;
    __syncthreads();
  }

  // B = sum over masked xv (fixed-order tree => deterministic)
  red[t] = msk ? xvv : 0.0f;
  __syncthreads();
  for (int s = 128; s > 0; s >>= 1) {
    if (t < s) red[t] += red[t + s];
    __syncthreads();
  }
  const float B = red[0];
  __syncthreads();

  // Each component root (lab == own index) gathers its mass deterministically.
  float Hloc = 0.0f;
  if (msk && lab[pos] == t) {
    float ssum = 0.0f;
    for (int q = 0; q < P; ++q) {
      const int qp = (q / SIDE + 1) * 16 + (q % SIDE + 1);
      if (lab[qp] == t) ssum += xvs[q];
    }
    const float pn = ssum / B;
    Hloc = -pn * logf(pn);
  }
  red[t] = Hloc;
  __syncthreads();
  for (int s = 128; s > 0; s >>= 1) {
    if (t < s) red[t] += red[t + s];
    __syncthreads();
  }
  if (t == 0) hsum[img] = red[0];
}

// ---------------------------------------------------------------------------
// Pass 3: deterministic final reduction of per-image entropies.
// ---------------------------------------------------------------------------
__global__ void final_reduce(const float* __restrict__ hsum,
                             float* __restrict__ out, int n) {
  __shared__ float red[256];
  const int t = threadIdx.x;
  float s = 0.0f;
  for (int i = t; i < n; i += 256) s += hsum[i];
  red[t] = s;
  __syncthreads();
  for (int st = 128; st > 0; st >>= 1) {
    if (t < st) red[t] += red[t + st];
    __syncthreads();
  }
  if (t == 0) out[0] = red[0] / (float)n;
}

extern "C" void kernel_launch(void* const* d_in, const int* in_sizes, int n_in,
                              void* d_out, int out_size, void* d_ws, size_t ws_size,
                              hipStream_t stream) {
  const float* dq = (const float*)d_in[0];
  const int n = in_sizes[0] / SS;         // 128 * 12 = 1536 images
  float* wsf = (float*)d_ws;
  float* means = wsf;                      // n floats
  float* hsum  = wsf + n;                  // n floats
  const int tiles = (n + 15) / 16;

  means_wmma<<<tiles, 32, 0, stream>>>(dq, means, n);
  blob_kernel<<<n, 256, 0, stream>>>(dq, means, hsum);
  final_reduce<<<1, 256, 0, stream>>>(hsum, (float*)d_out, n);
}